// Inference_and_Generation_85280870629440
// MI455X (gfx1250) — compile-verified
//
#include <hip/hip_runtime.h>
#include <math.h>

typedef float v2f __attribute__((ext_vector_type(2)));
typedef float v8f __attribute__((ext_vector_type(8)));

#define NW 32
#define NH 32
#define NCELL 1024
#define B 8
#define CF 64
#define W 256
#define H 256
#define K_OBJ 32
#define S 32
#define DZ 8
#define DZI 32
#define PDEC 2048

__device__ __forceinline__ float sigmoidf(float x) { return 1.0f / (1.0f + __expf(-x)); }

// ---------------- workspace layout (floats) ----------------
// score    : [B*1024]              off 0
// bxm      : [B*1024]              off 8192
// bym      : [B*1024]              off 16384
// bwm      : [B*1024]              off 24576
// bhm      : [B*1024]              off 32768
// cm       : [B*1024]              off 40960
// par_bx   : [256]  (m = k*8+b)    off 49152
// par_by   : [256]                 off 49408
// par_bw   : [256]                 off 49664
// par_bh   : [256]                 off 49920
// par_cf   : [256]                 off 50176
// pooled   : [256*64]              off 50432
// z        : [256*32]              off 66816
// small    : [256*2048]            off 75008
// total floats: 599296  (~2.4 MB)

#define OFF_SCORE  0
#define OFF_BXM    8192
#define OFF_BYM    16384
#define OFF_BWM    24576
#define OFF_BHM    32768
#define OFF_CM     40960
#define OFF_PBX    49152
#define OFF_PBY    49408
#define OFF_PBW    49664
#define OFF_PBH    49920
#define OFF_PCF    50176
#define OFF_POOL   50432
#define OFF_Z      66816
#define OFF_SMALL  75008

// ---------------- kernel 1: per-cell maps ----------------
__global__ void k1_cellmaps(const float* __restrict__ logit_map,
                            const float* __restrict__ zwhere_mu,
                            const float* __restrict__ w_zwhere,
                            const float* __restrict__ b_zwhere,
                            float* __restrict__ ws) {
    int idx = blockIdx.x * blockDim.x + threadIdx.x;   // 0..8191
    if (idx >= B * NCELL) return;
    int b    = idx / NCELL;
    int cell = idx % NCELL;
    int cw   = cell / NH;
    int ch   = cell % NH;

    float t[4];
    for (int o = 0; o < 4; ++o) {
        float acc = b_zwhere[o];
        for (int d = 0; d < DZ; ++d)
            acc += w_zwhere[o * DZ + d] * zwhere_mu[((b * DZ + d) * NCELL) + cell];
        t[o] = sigmoidf(acc);
    }
    float bx = ((float)cw + t[0]) * (float)(W / NW);
    float by = ((float)ch + t[1]) * (float)(H / NH);
    float bw = 16.0f + 48.0f * t[2];
    float bh = 16.0f + 48.0f * t[3];

    float p = sigmoidf(logit_map[b * NCELL + cell]);
    float c = (p > 0.5f) ? 1.0f : 0.0f;

    int o = b * NCELL + cell;
    ws[OFF_SCORE + o] = c + p;
    ws[OFF_BXM + o]   = bx;
    ws[OFF_BYM + o]   = by;
    ws[OFF_BWM + o]   = bw;
    ws[OFF_BHM + o]   = bh;
    ws[OFF_CM  + o]   = c;
}

// ---------------- kernel 2: per-batch top-32 + gather ----------------
__global__ void k2_topk(float* __restrict__ ws) {
    __shared__ float s[NCELL];
    __shared__ float rv[256];
    __shared__ int   ri[256];
    int b = blockIdx.x;
    int t = threadIdx.x;
    for (int q = 0; q < 4; ++q)
        s[t + 256 * q] = ws[OFF_SCORE + b * NCELL + t + 256 * q];
    __syncthreads();

    for (int k = 0; k < K_OBJ; ++k) {
        float bv = -1e30f; int bi = NCELL;
        for (int q = 0; q < 4; ++q) {
            int idx = t + 256 * q;
            float v = s[idx];
            if (v > bv || (v == bv && idx < bi)) { bv = v; bi = idx; }
        }
        rv[t] = bv; ri[t] = bi;
        __syncthreads();
        for (int stride = 128; stride >= 1; stride >>= 1) {
            if (t < stride) {
                if (rv[t + stride] > rv[t] ||
                    (rv[t + stride] == rv[t] && ri[t + stride] < ri[t])) {
                    rv[t] = rv[t + stride]; ri[t] = ri[t + stride];
                }
            }
            __syncthreads();
        }
        if (t == 0) {
            int cell = ri[0];
            int m = k * B + b;
            int o = b * NCELL + cell;
            ws[OFF_PBX + m] = ws[OFF_BXM + o];
            ws[OFF_PBY + m] = ws[OFF_BYM + o];
            ws[OFF_PBW + m] = ws[OFF_BWM + o];
            ws[OFF_PBH + m] = ws[OFF_BHM + o];
            ws[OFF_PCF + m] = ws[OFF_CM  + o];
            s[cell] = -2e30f;
        }
        __syncthreads();
    }
}

// ---------------- kernel 3: crop + mean pool ----------------
// one block per m = k*8+b ; 256 threads = 8 waves; wave owns 8 channels,
// lanes map to the y-axis of the sample grid (coalesced-ish), shuffle reduce.
__global__ void k3_croppool(const float* __restrict__ features,
                            float* __restrict__ ws) {
    __shared__ int   ix0[S], ix1[S], iy0[S], iy1[S];
    __shared__ float fwx[S], fvx[S], fwy[S], fvy[S];

    int m = blockIdx.x;          // k*8+b
    int b = m % B;
    int t = threadIdx.x;

    float bx = ws[OFF_PBX + m];
    float by = ws[OFF_PBY + m];
    float bw = ws[OFF_PBW + m];
    float bh = ws[OFF_PBH + m];

    if (t < S) {
        int i = t;
        float xs = bx - 0.5f * bw + ((float)i + 0.5f) * bw * (1.0f / (float)S) - 0.5f;
        float x0 = floorf(xs);
        fwx[i] = xs - x0;
        fvx[i] = (xs >= -1.0f && xs <= (float)W) ? 1.0f : 0.0f;
        int x0i = (int)x0;
        ix0[i] = min(max(x0i, 0), W - 1);
        ix1[i] = min(max(x0i + 1, 0), W - 1);
    } else if (t < 2 * S) {
        int j = t - S;
        float ysv = by - 0.5f * bh + ((float)j + 0.5f) * bh * (1.0f / (float)S) - 0.5f;
        float y0 = floorf(ysv);
        fwy[j] = ysv - y0;
        fvy[j] = (ysv >= -1.0f && ysv <= (float)H) ? 1.0f : 0.0f;
        int y0i = (int)y0;
        iy0[j] = min(max(y0i, 0), H - 1);
        iy1[j] = min(max(y0i + 1, 0), H - 1);
    }
    __syncthreads();

    int wave = t >> 5;
    int lane = t & 31;
    const float* Fb = features + (size_t)b * CF * W * H;

    int y0 = iy0[lane], y1 = iy1[lane];
    float wy = fwy[lane], vy = fvy[lane];

    for (int cc = 0; cc < 8; ++cc) {
        int c = wave * 8 + cc;
        const float* Fc = Fb + (size_t)c * W * H;
        float sum = 0.0f;
        for (int i = 0; i < S; ++i) {
            int x0 = ix0[i], x1 = ix1[i];
            float wx = fwx[i];
            float vv = fvx[i] * vy;
            float f00 = Fc[x0 * H + y0];
            float f01 = Fc[x0 * H + y1];
            float f10 = Fc[x1 * H + y0];
            float f11 = Fc[x1 * H + y1];
            float sx0 = (1.0f - wy) * f00 + wy * f01;
            float sx1 = (1.0f - wy) * f10 + wy * f11;
            sum += ((1.0f - wx) * sx0 + wx * sx1) * vv;
        }
        // fixed-order wave32 butterfly reduction (deterministic)
        for (int off = 16; off >= 1; off >>= 1)
            sum += __shfl_xor(sum, off, 32);
        if (lane == 0)
            ws[OFF_POOL + m * CF + c] = sum * (1.0f / (float)(S * S));
    }
}

// ---------------- kernel 4a: encoder GEMM (WMMA f32 16x16x4) ----------------
// z[m,o] = sum_c pooled[m,c] * w_enc[o,c] + b_enc[o]   for o < 32
// M=256, N=32, K=64 -> 16 mtiles x 2 ntiles = 32 wave tiles
__global__ void k4a_enc(const float* __restrict__ w_enc,
                        const float* __restrict__ b_enc,
                        float* __restrict__ ws) {
    int wid  = threadIdx.x >> 5;
    int lane = threadIdx.x & 31;
    int wt   = blockIdx.x * 8 + wid;      // 0..31
    int mtile = wt >> 1;
    int ntile = wt & 1;
    int khalf = lane >> 4;
    int mrow  = mtile * 16 + (lane & 15);
    int ncol  = ntile * 16 + (lane & 15);

    const float* pooled = ws + OFF_POOL;
    v8f acc = {};
    for (int kb = 0; kb < CF; kb += 4) {
        int k0 = kb + 2 * khalf;
        v2f a, bf;
        a.x  = pooled[mrow * CF + k0];
        a.y  = pooled[mrow * CF + k0 + 1];
        bf.x = w_enc[ncol * CF + k0];      // B[k][n] = w_enc[n][k]
        bf.y = w_enc[ncol * CF + k0 + 1];
        acc = __builtin_amdgcn_wmma_f32_16x16x4_f32(false, a, false, bf,
                                                    (short)0, acc, false, false);
    }
    float bias = b_enc[ncol];
    for (int v = 0; v < 8; ++v) {
        int m_out = mtile * 16 + v + 8 * khalf;
        ws[OFF_Z + m_out * DZI + ncol] = acc[v] + bias;
    }
}

// ---------------- kernel 4b: decoder GEMM (WMMA f32 16x16x4) + sigmoid ----
// small[m,p] = sigmoid(sum_d z[m,d] * w_dec[d,p] + b_dec[p])
// M=256, N=2048, K=32 -> 16 x 128 = 2048 wave tiles, 8 K-steps each
__global__ void k4b_dec(const float* __restrict__ w_dec,
                        const float* __restrict__ b_dec,
                        float* __restrict__ ws) {
    int wid  = threadIdx.x >> 5;
    int lane = threadIdx.x & 31;
    int wt   = blockIdx.x * 8 + wid;      // 0..2047
    int mtile = wt >> 7;                  // 0..15
    int ntile = wt & 127;                 // 0..127
    int khalf = lane >> 4;
    int mrow  = mtile * 16 + (lane & 15);
    int ncol  = ntile * 16 + (lane & 15);

    const float* z = ws + OFF_Z;
    v8f acc = {};
    for (int kb = 0; kb < DZI; kb += 4) {
        int k0 = kb + 2 * khalf;
        v2f a, bf;
        a.x  = z[mrow * DZI + k0];
        a.y  = z[mrow * DZI + k0 + 1];
        bf.x = w_dec[k0 * PDEC + ncol];
        bf.y = w_dec[(k0 + 1) * PDEC + ncol];
        acc = __builtin_amdgcn_wmma_f32_16x16x4_f32(false, a, false, bf,
                                                    (short)0, acc, false, false);
    }
    float bias = b_dec[ncol];
    for (int v = 0; v < 8; ++v) {
        int m_out = mtile * 16 + v + 8 * khalf;
        ws[OFF_SMALL + m_out * PDEC + ncol] = sigmoidf(acc[v] + bias);
    }
}

// ---------------- kernel 5: composite ----------------
__global__ void k5_composite(const float* __restrict__ bg,
                             const float* __restrict__ ws,
                             float* __restrict__ out) {
    __shared__ float Pbx[K_OBJ], Pby[K_OBJ], Pbw[K_OBJ], Pbh[K_OBJ], Pcf[K_OBJ];
    int t = threadIdx.x;
    int pix = blockIdx.x * blockDim.x + t;
    int b = pix / (W * H);
    int rem = pix % (W * H);
    int w = rem / H;
    int h = rem % H;

    if (t < K_OBJ) {
        int m = t * B + b;
        Pbx[t] = ws[OFF_PBX + m];
        Pby[t] = ws[OFF_PBY + m];
        Pbw[t] = ws[OFF_PBW + m];
        Pbh[t] = ws[OFF_PBH + m];
        Pcf[t] = ws[OFF_PCF + m];
    }
    __syncthreads();

    const float* smallp = ws + OFF_SMALL;
    float acc_mc = 0.0f;   // sum_k mc
    float acc_mi = 0.0f;   // sum_k mc * img
    float fw = (float)w + 0.5f;
    float fh = (float)h + 0.5f;

    for (int k = 0; k < K_OBJ; ++k) {
        float bw = Pbw[k], bh = Pbh[k];
        float u = (fw - (Pbx[k] - 0.5f * bw)) / bw * (float)S - 0.5f;
        float v = (fh - (Pby[k] - 0.5f * bh)) / bh * (float)S - 0.5f;
        if (u < -1.0f || u > (float)S || v < -1.0f || v > (float)S) continue;
        float x0 = floorf(u), y0 = floorf(v);
        float fx = u - x0, fy = v - y0;
        int x0i = (int)x0, y0i = (int)y0;
        int xa = min(max(x0i, 0), S - 1), xb = min(max(x0i + 1, 0), S - 1);
        int ya = min(max(y0i, 0), S - 1), yb = min(max(y0i + 1, 0), S - 1);
        const float* base = smallp + (size_t)(k * B + b) * PDEC;   // ch0 = mask
        float m00 = base[xa * S + ya], m01 = base[xa * S + yb];
        float m10 = base[xb * S + ya], m11 = base[xb * S + yb];
        float mval = (1.0f - fx) * ((1.0f - fy) * m00 + fy * m01)
                   +         fx  * ((1.0f - fy) * m10 + fy * m11);
        const float* basei = base + S * S;                          // ch1 = img
        float i00 = basei[xa * S + ya], i01 = basei[xa * S + yb];
        float i10 = basei[xb * S + ya], i11 = basei[xb * S + yb];
        float ival = (1.0f - fx) * ((1.0f - fy) * i00 + fy * i01)
                   +         fx  * ((1.0f - fy) * i10 + fy * i11);
        float mc = mval * Pcf[k];
        acc_mc += mc;
        acc_mi += mc * ival;
    }
    float denom = fmaxf(acc_mc, 1.0f);
    float mixsum = acc_mc / denom;
    out[pix] = acc_mi / denom + (1.0f - mixsum) * bg[pix];
}

extern "C" void kernel_launch(void* const* d_in, const int* in_sizes, int n_in,
                              void* d_out, int out_size, void* d_ws, size_t ws_size,
                              hipStream_t stream) {
    const float* logit_map = (const float*)d_in[1];
    const float* zwhere_mu = (const float*)d_in[2];
    const float* features  = (const float*)d_in[3];
    const float* bg        = (const float*)d_in[4];
    const float* w_zwhere  = (const float*)d_in[5];
    const float* b_zwhere  = (const float*)d_in[6];
    const float* w_enc     = (const float*)d_in[7];
    const float* b_enc     = (const float*)d_in[8];
    const float* w_dec     = (const float*)d_in[9];
    const float* b_dec     = (const float*)d_in[10];
    float* out = (float*)d_out;
    float* ws  = (float*)d_ws;

    k1_cellmaps<<<32, 256, 0, stream>>>(logit_map, zwhere_mu, w_zwhere, b_zwhere, ws);
    k2_topk<<<B, 256, 0, stream>>>(ws);
    k3_croppool<<<K_OBJ * B, 256, 0, stream>>>(features, ws);
    k4a_enc<<<4, 256, 0, stream>>>(w_enc, b_enc, ws);
    k4b_dec<<<256, 256, 0, stream>>>(w_dec, b_dec, ws);
    k5_composite<<<(B * W * H) / 256, 256, 0, stream>>>(bg, ws, out);
}